// subc_GNN_7765300871413
// MI455X (gfx1250) — compile-verified
//
#include <hip/hip_runtime.h>
#include <hip/hip_bf16.h>

typedef __attribute__((ext_vector_type(2))) float v2f;
typedef __attribute__((ext_vector_type(8))) float v8f;

#define WAVES_PER_WG 4
#define NSUB 16384
#define NBLK 1024

// One wave32 per 16-node subgraph. f32 WMMA (16x16x4) for both the dense
// layer GEMM and the block-diagonal propagation. Exploits the guaranteed
// block-diagonal adjacency: only the 16x16 diagonal blocks are read (1 MB
// instead of 1.07 GB -> removes the HBM bottleneck entirely).
//
// W0/W1 are staged once per workgroup into a fragment-major LDS layout so the
// inner-loop B operand is a single conflict-free ds_load_b64 per WMMA instead
// of two global_load_b32 per wave (LDS is the high-bandwidth path on CDNA5).
__global__ __launch_bounds__(32 * WAVES_PER_WG)
void subc_gnn_kernel(const int* __restrict__ node_types,
                     const float* __restrict__ node_feats,
                     const float* __restrict__ adj,
                     const float* __restrict__ W_cat,
                     const float* __restrict__ b_cat,
                     const float* __restrict__ W_num,
                     const float* __restrict__ b_num,
                     const float* __restrict__ W0,
                     const float* __restrict__ b0,
                     const float* __restrict__ W1,
                     const float* __restrict__ b1,
                     float* __restrict__ out)
{
    // Fragment-major W: [layer][kstep][ntile][lane][2] -> lane L reads banks
    // 2L,2L+1 as one 8B-aligned b64: all 64 banks hit exactly once.
    __shared__ float wf[2 * 16 * 4 * 64];            // 32 KB, workgroup-shared
    // Per-wave tiles, padded for conflict-free column access.
    __shared__ float x_lds[WAVES_PER_WG][16][68];    // x / y / z, 16 rows x 64 cols
    __shared__ float a_lds[WAVES_PER_WG][16][20];    // 16x16 adj block (80B rows, 16B aligned)
    __shared__ float inv_lds[WAVES_PER_WG][16];      // 1/(1+deg) per row

    const int lane  = threadIdx.x & 31;
    const int wave  = threadIdx.x >> 5;
    const int blk   = blockIdx.x * WAVES_PER_WG + wave;  // 0..1023
    const int base  = blk * 16;                           // first sub-node of segment

    float (*X)[68] = x_lds[wave];
    float (*A)[20] = a_lds[wave];
    float *INV     = inv_lds[wave];

    const int lhalf = lane >> 4;    // 0: lanes 0-15, 1: lanes 16-31
    const int lmod  = lane & 15;

    // ---- cooperative stage of W0/W1 fragments into LDS (one-time) ----
    for (int e = threadIdx.x; e < 2 * 16 * 4 * 32; e += 32 * WAVES_PER_WG) {
        int l  = e >> 11;          // layer
        int kk = (e >> 7) & 15;    // k-step
        int j  = (e >> 5) & 3;     // n-tile
        int ln = e & 31;           // lane slot
        int lh = ln >> 4, lm = ln & 15;
        const float* W = l ? W1 : W0;
        int row = 4 * kk + 2 * lh;
        int col = 16 * j + lm;
        wf[e * 2 + 0] = W[(size_t)row       * 64 + col];
        wf[e * 2 + 1] = W[(size_t)(row + 1) * 64 + col];
    }

    // ---- load this block's 16x16 adjacency tile as 2x float4 per lane ----
    {
        int i  = lane >> 1;              // row 0..15
        int c0 = (lane & 1) << 3;        // col 0 or 8
        const float4* src =
            reinterpret_cast<const float4*>(&adj[(size_t)(base + i) * NSUB + (size_t)(base + c0)]);
        float4 p0 = src[0];
        float4 p1 = src[1];
        *reinterpret_cast<float4*>(&A[i][c0])     = p0;
        *reinterpret_cast<float4*>(&A[i][c0 + 4]) = p1;
    }

    // ---- build input features x[16][64] (32 elems / lane) ----
    for (int t = lane; t < 16 * 64; t += 32) {
        int i = t >> 6, c = t & 63;
        int row = base + i;
        float v;
        if (c < 32) {
            v = W_cat[node_types[row] * 32 + c] + b_cat[c];
        } else {
            v = node_feats[row] * W_num[c - 32] + b_num[c - 32];
        }
        X[i][c] = v;
    }
    __syncthreads();   // wf is cross-wave; also orders per-wave A/X staging

    // ---- inv_deg from column sums of the block (symmetric => row degrees) ----
    if (lane < 16) {
        float s = 1.0f;
        #pragma unroll
        for (int i = 0; i < 16; ++i) s += A[i][lane];
        INV[lane] = 1.0f / s;
    }
    __builtin_amdgcn_wave_barrier();   // per-wave data; DS ops are in-order per wave

    const float* bl[2] = { b0, b1 };

    v8f acc[4];

    #pragma unroll
    for (int layer = 0; layer < 2; ++layer) {
        const float* bb = bl[layer];

        // ---- y = x @ W + b : 4 N-tiles, K=64 in steps of 4 ----
        #pragma unroll
        for (int j = 0; j < 4; ++j) {
            float bias = bb[16 * j + lmod];          // bias depends on column only
            v8f c;
            #pragma unroll
            for (int r = 0; r < 8; ++r) c[r] = bias;
            #pragma unroll
            for (int k = 0; k < 64; k += 4) {
                v2f a, b;
                // A fragment (16x4 f32): lanes 0-15 -> K=k,k+1 ; lanes 16-31 -> K=k+2,k+3
                a.x = X[lmod][k + 2 * lhalf];
                a.y = X[lmod][k + 2 * lhalf + 1];
                // B fragment: one conflict-free 8B LDS load per lane
                b = *reinterpret_cast<const v2f*>(
                        &wf[(((layer * 16) + (k >> 2)) * 4 + j) * 64 + lane * 2]);
                c = __builtin_amdgcn_wmma_f32_16x16x4_f32(
                        false, a, false, b, (short)0, c, false, false);
            }
            acc[j] = c;
        }
        __builtin_amdgcn_wave_barrier();

        // ---- spill y to LDS so it can be re-read as a B operand ----
        #pragma unroll
        for (int j = 0; j < 4; ++j)
            #pragma unroll
            for (int r = 0; r < 8; ++r)
                X[r + 8 * lhalf][16 * j + lmod] = acc[j][r];
        __builtin_amdgcn_wave_barrier();

        // ---- z = relu((y + A_b @ y) * inv_deg) : C seeded with y ----
        #pragma unroll
        for (int j = 0; j < 4; ++j) {
            v8f c = acc[j];                          // (I + A) x : start from y
            #pragma unroll
            for (int k = 0; k < 16; k += 4) {
                v2f a, b;
                a.x = A[lmod][k + 2 * lhalf];
                a.y = A[lmod][k + 2 * lhalf + 1];
                b.x = X[k + 2 * lhalf][16 * j + lmod];
                b.y = X[k + 2 * lhalf + 1][16 * j + lmod];
                c = __builtin_amdgcn_wmma_f32_16x16x4_f32(
                        false, a, false, b, (short)0, c, false, false);
            }
            #pragma unroll
            for (int r = 0; r < 8; ++r) {
                float z = c[r] * INV[r + 8 * lhalf]; // row scaling
                acc[j][r] = z > 0.0f ? z : 0.0f;     // relu
            }
        }
        __builtin_amdgcn_wave_barrier();

        // ---- write z back for next layer's A-operand reads ----
        #pragma unroll
        for (int j = 0; j < 4; ++j)
            #pragma unroll
            for (int r = 0; r < 8; ++r)
                X[r + 8 * lhalf][16 * j + lmod] = acc[j][r];
        __builtin_amdgcn_wave_barrier();
    }

    // ---- segment-sum readout: sum the 16 rows of each column ----
    #pragma unroll
    for (int j = 0; j < 4; ++j) {
        float s = 0.0f;
        #pragma unroll
        for (int r = 0; r < 8; ++r) s += acc[j][r];  // rows r+8*lhalf
        s += __shfl_xor(s, 16);                      // combine the two row-halves
        if (lhalf == 0)
            out[(size_t)blk * 64 + 16 * j + lmod] = s;
    }
}

extern "C" void kernel_launch(void* const* d_in, const int* in_sizes, int n_in,
                              void* d_out, int out_size, void* d_ws, size_t ws_size,
                              hipStream_t stream) {
    (void)in_sizes; (void)n_in; (void)d_ws; (void)ws_size; (void)out_size;
    const int*   node_types = (const int*)  d_in[0];
    const float* node_feats = (const float*)d_in[1];
    const float* all_adj    = (const float*)d_in[2];
    // d_in[3] = segment_ids: segments are contiguous blocks of 16 by construction
    const float* W_cat      = (const float*)d_in[4];
    const float* b_cat      = (const float*)d_in[5];
    const float* W_num      = (const float*)d_in[6];
    const float* b_num      = (const float*)d_in[7];
    const float* W0         = (const float*)d_in[8];
    const float* b0         = (const float*)d_in[9];
    const float* W1         = (const float*)d_in[10];
    const float* b1         = (const float*)d_in[11];
    float* out              = (float*)d_out;

    dim3 grid(NBLK / WAVES_PER_WG);        // 256 workgroups
    dim3 block(32 * WAVES_PER_WG);         // 4 wave32 per workgroup
    subc_gnn_kernel<<<grid, block, 0, stream>>>(
        node_types, node_feats, all_adj, W_cat, b_cat, W_num, b_num,
        W0, b0, W1, b1, out);
}